// PositionalEncoding_19318762898057
// MI455X (gfx1250) — compile-verified
//
#include <hip/hip_runtime.h>
#include <hip/hip_bf16.h>

// ---------------------------------------------------------------------------
// PositionalEncoding gather, MI455X (gfx1250, wave32).
//
// Shapes (fixed by reference):
//   routing  [16, 4096] int32 (values 0/1)
//   pos_embed[4096, 512] f32                      (8 MB -> L2-resident)
//   node_pos [16, 1024], edge_pos [16, 2048]      (compacted indices; ws)
//   out = node_pe [16, 2048, 512] ++ edge_pe [16, 10240, 512]  (f32, 403 MB)
//
// Roofline: zero useful FLOPs; 403 MB of output -> write-bandwidth bound,
// floor ~= 403 MB / 23.3 TB/s ~= 17 us. Strategy:
//   * compact kernel also bulk-prefetches pos_embed into L2
//     (global_prefetch_b8, fire-and-forget) so gather reads are L2 hits,
//   * gather blocks read each 2 KB row ONCE via global_load_b128 and fan it
//     out to the 2x / 5x repeated destination rows from registers,
//   * all output stores are non-temporal (global_store_b128 th:NT) so the
//     403 MB stream never evicts pos_embed / indices from L2.
// ---------------------------------------------------------------------------

typedef float v4f __attribute__((ext_vector_type(4)));

#define B_DIM       16
#define T_DIM       4096
#define D_DIM       512
#define D4          128          // 512 floats / 4 per float4
#define N_NODES     1024
#define N_EDGES     2048
#define NODE_REP    2
#define EDGE_REP    5

// -------------------------------------------------------------------
// Kernel 1: per-batch-row stream compaction via LDS scan, plus L2
// warm-up prefetch of the embedding table.
// One block (256 threads) per batch row; 16 routing elements per thread.
// rank_among_ones(t) = t - rank_among_zeros(t), so a single zero-count
// scan serves both node and edge compaction.
// Slots never written stay 0, which equals the reference's max(-1, 0).
// -------------------------------------------------------------------
__global__ __launch_bounds__(256)
void pe_compact_kernel(const int* __restrict__ routing,
                       const float* __restrict__ pos_embed,
                       int* __restrict__ node_pos,
                       int* __restrict__ edge_pos) {
    const int b   = blockIdx.x;
    const int tid = threadIdx.x;                 // 0..255
    const int* r  = routing + b * T_DIM;
    int* np = node_pos + b * N_NODES;
    int* ep = edge_pos + b * N_EDGES;

    // ---- L2 warm-up: 16 blocks x 256 threads == 4096 threads, one 2 KB
    // embedding row per thread, prefetched at 128 B granularity.
    // global_prefetch_b8 is fire-and-forget (no LOADcnt), overlapping the
    // scan below and the gather kernel's launch.
    {
        const char* row = (const char*)(pos_embed +
                          ((size_t)b * 256 + tid) * D_DIM);
#pragma unroll
        for (int k = 0; k < 16; ++k)
            __builtin_prefetch(row + k * 128, /*rw=*/0, /*locality=*/3);
    }

    // Pre-fill with 0 (== gather index for invalid slots).
    for (int i = tid; i < N_NODES; i += 256) np[i] = 0;
    for (int i = tid; i < N_EDGES; i += 256) ep[i] = 0;

    __shared__ int s_cnt[256];

    int vals[16];
    int zc = 0;
#pragma unroll
    for (int k = 0; k < 16; ++k) {
        vals[k] = r[tid * 16 + k];
        zc += (vals[k] == 0);
    }
    s_cnt[tid] = zc;
    __syncthreads();   // also orders the zero-fill above vs. scatter below

    // Hillis-Steele inclusive scan over 256 partial counts (8 steps).
#pragma unroll
    for (int off = 1; off < 256; off <<= 1) {
        int mine = s_cnt[tid];
        int add  = (tid >= off) ? s_cnt[tid - off] : 0;
        __syncthreads();
        s_cnt[tid] = mine + add;
        __syncthreads();
    }
    const int excl = (tid == 0) ? 0 : s_cnt[tid - 1];

    // Scatter positions.
    int run = excl;                              // zeros strictly before t
#pragma unroll
    for (int k = 0; k < 16; ++k) {
        const int t = tid * 16 + k;
        if (vals[k] == 0) {
            if (run < N_NODES) np[run] = t;
            ++run;
        } else {
            const int e = t - run;               // ones strictly before t
            if (e < N_EDGES) ep[e] = t;
        }
    }
}

// -------------------------------------------------------------------
// Kernel 2: fused node+edge gather with repeat fan-out.
// 128 threads per block (4 wave32 waves); one block per compacted slot.
// Each thread: one b128 load of the embedding row (L2 hit), then 2 or 5
// b128 non-temporal stores of the repeated rows.
// -------------------------------------------------------------------
__global__ __launch_bounds__(128)
void pe_gather_kernel(const float* __restrict__ pos_embed,
                      const int* __restrict__ node_pos,
                      const int* __restrict__ edge_pos,
                      float* __restrict__ out) {
    const int tid = threadIdx.x;                 // 0..127 -> float4 lane in row
    int blk = blockIdx.x;
    const v4f* pe = (const v4f*)pos_embed;
    v4f* o4 = (v4f*)out;

    if (blk < B_DIM * N_NODES) {
        const int b = blk >> 10;                 // /1024
        const int s = blk & (N_NODES - 1);
        const int tok = node_pos[b * N_NODES + s];
        const v4f v = pe[(size_t)tok * D4 + tid];
        v4f* dst = o4 + ((size_t)b * (N_NODES * NODE_REP) + (size_t)s * NODE_REP) * D4 + tid;
#pragma unroll
        for (int rep = 0; rep < NODE_REP; ++rep)
            __builtin_nontemporal_store(v, dst + (size_t)rep * D4);
    } else {
        blk -= B_DIM * N_NODES;
        const int b = blk >> 11;                 // /2048
        const int s = blk & (N_EDGES - 1);
        const int tok = edge_pos[b * N_EDGES + s];
        const v4f v = pe[(size_t)tok * D4 + tid];
        const size_t edge_base = (size_t)B_DIM * (N_NODES * NODE_REP) * D4; // after node_pe
        v4f* dst = o4 + edge_base
                      + ((size_t)b * (N_EDGES * EDGE_REP) + (size_t)s * EDGE_REP) * D4 + tid;
#pragma unroll
        for (int rep = 0; rep < EDGE_REP; ++rep)
            __builtin_nontemporal_store(v, dst + (size_t)rep * D4);
    }
}

extern "C" void kernel_launch(void* const* d_in, const int* in_sizes, int n_in,
                              void* d_out, int out_size, void* d_ws, size_t ws_size,
                              hipStream_t stream) {
    (void)in_sizes; (void)n_in; (void)out_size; (void)ws_size;

    const int*   routing   = (const int*)d_in[0];   // [16, 4096]
    // d_in[1] = max_nodes (1024), d_in[2] = max_edges (2048): compile-time here.
    const float* pos_embed = (const float*)d_in[3]; // [4096, 512]

    int* node_pos = (int*)d_ws;                     // 16*1024 ints
    int* edge_pos = node_pos + B_DIM * N_NODES;     // 16*2048 ints  (192 KB total)

    pe_compact_kernel<<<B_DIM, 256, 0, stream>>>(routing, pos_embed,
                                                 node_pos, edge_pos);

    const int nblocks = B_DIM * N_NODES + B_DIM * N_EDGES;  // 49152
    pe_gather_kernel<<<nblocks, 128, 0, stream>>>(pos_embed, node_pos, edge_pos,
                                                  (float*)d_out);
}